// SecSolver_28724741276004
// MI455X (gfx1250) — compile-verified
//
#include <hip/hip_runtime.h>

#define NSWEEP 12

typedef float v2f __attribute__((ext_vector_type(2)));
typedef float v8f __attribute__((ext_vector_type(8)));

__device__ __forceinline__ void lds_fence() { asm volatile("" ::: "memory"); }

// Stable ascending insertion argsort of 16 floats.
// Sorts v in place (v becomes take_along(energies, eb2e)); ord = eb2e.
__device__ __forceinline__ void argsort16(float* v, int* ord) {
#pragma unroll
  for (int k = 0; k < 16; ++k) ord[k] = k;
  for (int i = 1; i < 16; ++i) {
    float kv = v[i]; int ki = ord[i];
    int j = i - 1;
    while (j >= 0 && v[j] > kv) { v[j + 1] = v[j]; ord[j + 1] = ord[j]; --j; }
    v[j + 1] = kv; ord[j + 1] = ki;
  }
}

// energies tile: E[m0+r][k] = rf[m0+r]*gz[k] + evals[k] via V_WMMA_F32_16X16X4_F32.
// A (16x4): lanes 0-15 carry K=0 (rf), lanes 16-31 carry K=2 (zero).
// B (4x16): lanes 0-15 carry K=0 row (gz), lanes 16-31 K=2 row (zero).
// C/D (16x16): VGPR v -> row v (lanes 0-15) / row v+8 (lanes 16-31), col = lane&15.
__device__ __forceinline__ void energies_tiles_wmma(
    const float* rfv, const float* evl, const float* gzl,
    float (*eE)[17], int wave, int lane) {
  const bool lo = lane < 16;
  const int n = lane & 15;
  v2f bmat; bmat.x = lo ? gzl[n] : 0.0f; bmat.y = 0.0f;
  const float e = evl[n];
  v8f cm;
#pragma unroll
  for (int v = 0; v < 8; ++v) cm[v] = e;
#pragma unroll
  for (int tt = 0; tt < 2; ++tt) {
    const int m0 = (wave * 2 + tt) * 16;
    v2f amat; amat.x = lo ? rfv[m0 + n] : 0.0f; amat.y = 0.0f;
    v8f d = __builtin_amdgcn_wmma_f32_16x16x4_f32(
        false, amat, false, bmat, (short)0, cm, false, false);
#pragma unroll
    for (int v = 0; v < 8; ++v) {
      const int m = m0 + v + (lo ? 0 : 8);
      eE[m][n] = d[v];
    }
  }
}

__global__ void init_union_kernel(int* u) { u[threadIdx.x] = 0; }

// ---------------- Kernel 1: Jacobi eigh + gz_eigen + B_res + union --------
__global__ void __launch_bounds__(256)
eigh_kernel(const float* __restrict__ F, const float* __restrict__ Gz,
            const float* __restrict__ Blo, const float* __restrict__ Bhi,
            const float* __restrict__ freqp,
            float* __restrict__ wevals, float* __restrict__ wgz,
            float* __restrict__ wevecs, float* __restrict__ wbres,
            int* __restrict__ wunion) {
  __shared__ float A[8][16][17];
  __shared__ float Vv[8][16][17];
  __shared__ float gzd[8][16];
  __shared__ float dia[8][16];
  __shared__ float evs[8][16];
  __shared__ float gzE[8][16];

  const int w = threadIdx.x >> 5;
  const int lane = threadIdx.x & 31;
  const int b = blockIdx.x * 8 + w;

  float (*Aw)[17] = A[w];
  float (*Vw)[17] = Vv[w];

#pragma unroll
  for (int r = 0; r < 8; ++r) {
    const int idx = r * 32 + lane;
    const int i = idx >> 4, j = idx & 15;
    Aw[i][j] = F[(size_t)b * 256 + idx];
    Vw[i][j] = (i == j) ? 1.0f : 0.0f;
  }
  if (lane < 16) gzd[w][lane] = Gz[(size_t)b * 256 + lane * 16 + lane];
  lds_fence();

  if (lane < 16) {
    const int j = lane;
    for (int sweep = 0; sweep < NSWEEP; ++sweep) {
      for (int p = 0; p < 15; ++p) {
        for (int q = p + 1; q < 16; ++q) {
          const float apq = Aw[p][q];
          if (fabsf(apq) > 1e-13f) {
            const float app = Aw[p][p];
            const float aqq = Aw[q][q];
            const float tau = (aqq - app) / (2.0f * apq);
            const float t =
                ((tau >= 0.0f) ? 1.0f : -1.0f) /
                (fabsf(tau) + sqrtf(1.0f + tau * tau));
            const float c = 1.0f / sqrtf(1.0f + t * t);
            const float s = t * c;
            // row rotation (A' = J^T A)
            const float rp = Aw[p][j], rq = Aw[q][j];
            lds_fence();
            Aw[p][j] = c * rp - s * rq;
            Aw[q][j] = s * rp + c * rq;
            lds_fence();
            // column rotation (A'' = A' J)
            const float cp = Aw[j][p], cq = Aw[j][q];
            lds_fence();
            Aw[j][p] = c * cp - s * cq;
            Aw[j][q] = s * cp + c * cq;
            lds_fence();
            // accumulate eigenvectors V <- V J
            const float vp = Vw[j][p], vq = Vw[j][q];
            Vw[j][p] = c * vp - s * vq;
            Vw[j][q] = s * vp + c * vq;
            lds_fence();
          }
        }
      }
    }
    dia[w][j] = Aw[j][j];
  }
  lds_fence();

  // stable ascending rank sort of eigenvalues; permute eigenvector columns.
  if (lane < 16) {
    const float di = dia[w][lane];
    int r = 0;
#pragma unroll
    for (int jj = 0; jj < 16; ++jj) {
      const float dj = dia[w][jj];
      r += (int)((dj < di) || (dj == di && jj < lane));
    }
    evs[w][r] = di;
    for (int row = 0; row < 16; ++row) Aw[row][r] = Vw[row][lane];  // Aw := sorted evecs
  }
  lds_fence();
  // gz_eigen[i] = sum_j evecs[i][j]^2 * gz_diag[j]   (einsum 'bij,bj,bij->bi')
  if (lane < 16) {
    float acc = 0.0f;
#pragma unroll
    for (int jj = 0; jj < 16; ++jj) {
      const float v = Aw[lane][jj];
      acc += v * v * gzd[w][jj];
    }
    gzE[w][lane] = acc;
  }
  lds_fence();

  if (lane < 16) {
    wevals[(size_t)b * 16 + lane] = evs[w][lane];
    wgz[(size_t)b * 16 + lane] = gzE[w][lane];
  }
#pragma unroll
  for (int r = 0; r < 8; ++r) {
    const int idx = r * 32 + lane;
    wevecs[(size_t)b * 256 + idx] = Aw[idx >> 4][idx & 15];
  }

  const float f0 = freqp[0];
  const float bl = Blo[b], bh = Bhi[b];
#pragma unroll
  for (int r = 0; r < 8; ++r) {
    const int p = r * 32 + lane;
    const int i = p >> 4, jj = p & 15;
    const float dE = evs[w][i] - evs[w][jj];
    const float dgz = gzE[w][i] - gzE[w][jj];
    const float br = (f0 - dE) / (dgz + 1e-8f);
    wbres[(size_t)b * 256 + p] = br;
    const bool inr = (br >= bl) && (br <= bh) && (fabsf(dgz) > 1e-8f);
    if (inr) atomicOr(&wunion[p], 1);
  }
}

// ---------------- Kernel 2: compaction + WMMA energies + pid/firstpos -----
__global__ void __launch_bounds__(256)
pairs_kernel(const float* __restrict__ wbres, const int* __restrict__ wunion,
             const float* __restrict__ Blo, const float* __restrict__ Bhi,
             const float* __restrict__ wevals, const float* __restrict__ wgz,
             float* __restrict__ wrf, int* __restrict__ wfirst,
             unsigned char* __restrict__ womask,
             unsigned char* __restrict__ wmflat) {
  __shared__ float bresL[256];
  __shared__ int sc[256];
  __shared__ int sidx[256];
  __shared__ float rfL[256];
  __shared__ float evl[16];
  __shared__ float gzl[16];
  __shared__ float eE[256][17];
  __shared__ int fp[256];

  const int m = threadIdx.x;
  const int b = blockIdx.x;
  const int lane = m & 31;
  const int wave = m >> 5;

  const float br = wbres[(size_t)b * 256 + m];
  bresL[m] = br;
  if (m < 16) {
    evl[m] = wevals[(size_t)b * 16 + m];
    gzl[m] = wgz[(size_t)b * 16 + m];
  }
  const float bl = Blo[b], bh = Bhi[b];
  const int valid = ((br >= bl) && (br <= bh) && (wunion[m] != 0)) ? 1 : 0;
  sc[m] = valid;
  __syncthreads();
  for (int off = 1; off < 256; off <<= 1) {  // inclusive scan
    const int t = (m >= off) ? sc[m - off] : 0;
    __syncthreads();
    sc[m] += t;
    __syncthreads();
  }
  const int total = sc[255];
  if (valid) sidx[sc[m] - 1] = m;  // sorted valid positions (== jnp.sort(sent))
  __syncthreads();
  const bool omask = m < total;
  const int idx_safe = omask ? sidx[m] : 0;
  const float rfv = omask ? bresL[idx_safe] : 0.0f;
  rfL[m] = rfv;
  womask[(size_t)b * 256 + m] = omask ? 1 : 0;
  wrf[(size_t)b * 256 + m] = rfv;
  __syncthreads();

  energies_tiles_wmma(rfL, evl, gzl, eE, wave, lane);
  __syncthreads();

  float vals[16]; int ord[16];
#pragma unroll
  for (int k = 0; k < 16; ++k) vals[k] = eE[m][k];
  argsort16(vals, ord);          // ord = eb2e
  int inv[16];
#pragma unroll
  for (int k = 0; k < 16; ++k) inv[ord[k]] = k;  // e2eb
  const int i_idx = idx_safe & 15;   // high_idx (col)
  const int j_idx = idx_safe >> 4;   // low_idx (row)
  const int pid = inv[i_idx] * 16 + inv[j_idx];  // lvl_down*K + lvl_up

  fp[m] = 256;
  __syncthreads();
  atomicMin(&fp[pid], m);            // first occurrence per dense key
  __syncthreads();
  wfirst[(size_t)b * 256 + m] = fp[m];
  wmflat[(size_t)b * 256 + m] = (fp[m] < 256) ? 1 : 0;
}

// ---------------- Kernel 3: global cumsum masked-scatter ------------------
__global__ void __launch_bounds__(1024)
scan_kernel(const unsigned char* __restrict__ wmflat,
            const unsigned char* __restrict__ womask,
            float* __restrict__ outmask, int N) {
  __shared__ int sc[1024];
  const int tid = threadIdx.x;
  int carry = 0;
  for (int base = 0; base < N; base += 1024) {
    const int x = base + tid;
    const int ms = (int)wmflat[x];
    sc[tid] = ms;
    __syncthreads();
    for (int off = 1; off < 1024; off <<= 1) {
      const int t = (tid >= off) ? sc[tid - off] : 0;
      __syncthreads();
      sc[tid] += t;
      __syncthreads();
    }
    const int incl = sc[tid];
    float v = 0.0f;
    if (ms) {
      int r = carry + incl - 1;  // rank of this True among all Trues
      if (r < 0) r = 0;
      if (r >= N) r = N - 1;
      v = womask[r] ? 1.0f : 0.0f;
    }
    outmask[x] = v;
    carry += sc[1023];
    __syncthreads();
  }
}

// ---------------- Kernel 4: finalize outputs ------------------------------
__global__ void __launch_bounds__(256)
final_kernel(const float* __restrict__ wevals, const float* __restrict__ wgz,
             const float* __restrict__ wevecs, const float* __restrict__ wrf,
             const int* __restrict__ wfirst,
             const unsigned char* __restrict__ wmflat,
             const float* __restrict__ maskmerged,
             float* __restrict__ vup, float* __restrict__ vdn,
             float* __restrict__ lup, float* __restrict__ ldn,
             float* __restrict__ rfm_out, float* __restrict__ en_out) {
  __shared__ float rfmL[256];
  __shared__ float evl[16];
  __shared__ float gzl[16];
  __shared__ float Ve[16][17];
  __shared__ float eE[256][17];

  const int kk = threadIdx.x;
  const int b = blockIdx.x;
  const int lane = kk & 31;
  const int wave = kk >> 5;

  if (kk < 16) {
    evl[kk] = wevals[(size_t)b * 16 + kk];
    gzl[kk] = wgz[(size_t)b * 16 + kk];
  }
  Ve[kk >> 4][kk & 15] = wevecs[(size_t)b * 256 + kk];

  const int fpv = wfirst[(size_t)b * 256 + kk];
  const int col = wmflat[(size_t)b * 256 + kk] ? fpv : 0;
  const float rfm = wrf[(size_t)b * 256 + col];
  rfmL[kk] = rfm;
  rfm_out[(size_t)b * 256 + kk] = rfm;
  if (b == 0) {
    lup[kk] = (float)(kk & 15);   // lvl_up_out  = arange(M) % K
    ldn[kk] = (float)(kk >> 4);   // lvl_down_out = arange(M) // K
  }
  __syncthreads();

  energies_tiles_wmma(rfmL, evl, gzl, eE, wave, lane);
  __syncthreads();

  float vals[16]; int ord[16];
#pragma unroll
  for (int k = 0; k < 16; ++k) vals[k] = eE[kk][k];
  argsort16(vals, ord);  // vals -> sorted energies (out_energies), ord = eb2e_m

  const int cdown = ord[kk >> 4];
  const int cup = ord[kk & 15];
  const float mm = maskmerged[(size_t)b * 256 + kk];
  const size_t rowoff = ((size_t)b * 256 + kk) * 16;
#pragma unroll
  for (int k = 0; k < 16; ++k) en_out[rowoff + k] = vals[k];
#pragma unroll
  for (int i = 0; i < 16; ++i) {
    vdn[rowoff + i] = Ve[i][cdown] * mm;
    vup[rowoff + i] = Ve[i][cup] * mm;
  }
}

extern "C" void kernel_launch(void* const* d_in, const int* in_sizes, int n_in,
                              void* d_out, int out_size, void* d_ws,
                              size_t ws_size, hipStream_t stream) {
  const float* F = (const float*)d_in[0];
  const float* Gz = (const float*)d_in[1];
  const float* Blo = (const float*)d_in[2];
  const float* Bhi = (const float*)d_in[3];
  const float* freq = (const float*)d_in[4];
  const int B = in_sizes[2];  // batch count from B_low

  char* ws = (char*)d_ws;
  size_t off = 0;
  int* wunion = (int*)(ws + off);            off += 256 * sizeof(int);
  float* wevals = (float*)(ws + off);        off += (size_t)B * 16 * sizeof(float);
  float* wgz = (float*)(ws + off);           off += (size_t)B * 16 * sizeof(float);
  float* wevecs = (float*)(ws + off);        off += (size_t)B * 256 * sizeof(float);
  float* wbres = (float*)(ws + off);         off += (size_t)B * 256 * sizeof(float);
  float* wrf = (float*)(ws + off);           off += (size_t)B * 256 * sizeof(float);
  int* wfirst = (int*)(ws + off);            off += (size_t)B * 256 * sizeof(int);
  unsigned char* womask = (unsigned char*)(ws + off); off += (size_t)B * 256;
  unsigned char* wmflat = (unsigned char*)(ws + off); off += (size_t)B * 256;

  float* out = (float*)d_out;
  const size_t BMK = (size_t)B * 256 * 16;
  float* o_vup = out;
  float* o_vdn = out + BMK;
  float* o_lup = out + 2 * BMK;
  float* o_ldn = out + 2 * BMK + 256;
  float* o_rf = out + 2 * BMK + 512;
  float* o_en = o_rf + (size_t)B * 256;
  float* o_mm = o_en + BMK;

  init_union_kernel<<<1, 256, 0, stream>>>(wunion);
  eigh_kernel<<<B / 8, 256, 0, stream>>>(F, Gz, Blo, Bhi, freq, wevals, wgz,
                                         wevecs, wbres, wunion);
  pairs_kernel<<<B, 256, 0, stream>>>(wbres, wunion, Blo, Bhi, wevals, wgz,
                                      wrf, wfirst, womask, wmflat);
  scan_kernel<<<1, 1024, 0, stream>>>(wmflat, womask, o_mm, B * 256);
  final_kernel<<<B, 256, 0, stream>>>(wevals, wgz, wevecs, wrf, wfirst, wmflat,
                                      o_mm, o_vup, o_vdn, o_lup, o_ldn, o_rf,
                                      o_en);
}